// SequentiaSizeLoss_40973988004466
// MI455X (gfx1250) — compile-verified
//
#include <hip/hip_runtime.h>
#include <hip/hip_bf16.h>
#include <math.h>

// Problem constants (fixed by the reference setup_inputs)
#define BB 8
#define TT 16
#define HH 512
#define WW 512
#define FRAME_ELEMS (HH * WW)                 // 262144
#define FRAME_VEC4  (FRAME_ELEMS / 4)         // 65536
#define BLOCKS_PER_FRAME 32
#define CHUNK_VEC4  (FRAME_VEC4 / BLOCKS_PER_FRAME)  // 2048
#define THREADS 256
#define ITERS (CHUNK_VEC4 / THREADS)          // 8 (uniform, compile-time)
#define NFRAMES (BB * TT)                     // 128
#define NTOT (8.0 * 16.0 * 512.0 * 512.0)     // 33554432 elements

// Workspace layout (floats)
#define WS_BCE 0
#define WS_PT  1      // 8 entries
#define WS_P   9      // 8 entries
#define WS_T   17     // 8 entries
#define WS_S   32     // 128 entries (per-frame sums of raw x)
#define WS_FLOATS 160

typedef __attribute__((ext_vector_type(2))) float v2f;
typedef __attribute__((ext_vector_type(4))) float f32x4;
typedef __attribute__((ext_vector_type(8))) float v8f;

// ds_swizzle group-of-32 XOR patterns: offset = (xor<<10) | or(0)<<5 | and_mask(0x1F)
#define SWZ_XOR8 0x201F
#define SWZ_XOR4 0x101F
#define SWZ_XOR2 0x081F
#define SWZ_XOR1 0x041F

__device__ __forceinline__ float swz_add(float r, const int imm_is_handled_below) {
    return r; // placeholder (not used)
}

// Exact fp32 wave32 reduction through the CDNA5 matrix pipe.
// A = ones (16x4), B = {lane partial, 0} (4x16), C = 0.
// D[m,n] = sum_k B[k,n]  =>  every lane's d[0] holds acc[lane%16] + acc[lane%16+16],
// replicated across rows. XOR-fold the 16 column sums with ds_swizzle (masks
// 8/4/2/1 preserve the mod-16 replication group). All fp32 FMA, RNE — exact.
__device__ __forceinline__ float wave_reduce_wmma(float v) {
    v2f a; a.x = 1.0f; a.y = 1.0f;
    v2f b; b.x = v;    b.y = 0.0f;
    v8f c = {};
    c = __builtin_amdgcn_wmma_f32_16x16x4_f32(
            /*neg_a=*/false, a, /*neg_b=*/false, b,
            /*c_mod=*/(short)0, c, /*reuse_a=*/false, /*reuse_b=*/false);
    float r = c[0];
    r += __int_as_float(__builtin_amdgcn_ds_swizzle(__float_as_int(r), SWZ_XOR8));
    r += __int_as_float(__builtin_amdgcn_ds_swizzle(__float_as_int(r), SWZ_XOR4));
    r += __int_as_float(__builtin_amdgcn_ds_swizzle(__float_as_int(r), SWZ_XOR2));
    r += __int_as_float(__builtin_amdgcn_ds_swizzle(__float_as_int(r), SWZ_XOR1));
    return r;
}

__global__ __launch_bounds__(THREADS) void zero_ws_kernel(float* __restrict__ ws) {
    int i = threadIdx.x;
    if (i < WS_FLOATS) ws[i] = 0.0f;
}

__global__ __launch_bounds__(THREADS)
void bce_dice_partial_kernel(const f32x4* __restrict__ in4,
                             const f32x4* __restrict__ tg4,
                             float* __restrict__ ws) {
    const int blk   = blockIdx.x;
    const int frame = blk / BLOCKS_PER_FRAME;      // 0..127  (b*16 + t)
    const int sub   = blk % BLOCKS_PER_FRAME;
    const int b     = frame / TT;
    const long long base = (long long)frame * FRAME_VEC4 + (long long)sub * CHUNK_VEC4;

    float a_bce = 0.0f, a_p = 0.0f, a_t = 0.0f, a_pt = 0.0f, a_x = 0.0f;

    // Uniform trip count (8) -> no exec-masked loop, full unroll, batched b128 loads.
#pragma unroll
    for (int it = 0; it < ITERS; ++it) {
        const int i = it * THREADS + threadIdx.x;
        f32x4 xv = __builtin_nontemporal_load(&in4[base + i]);
        f32x4 tv = __builtin_nontemporal_load(&tg4[base + i]);
        float xs[4] = {xv.x, xv.y, xv.z, xv.w};
        float ts[4] = {tv.x, tv.y, tv.z, tv.w};
#pragma unroll
        for (int k = 0; k < 4; ++k) {
            float x = xs[k];
            float t = ts[k];
            float e   = __expf(-fabsf(x));            // shared between BCE and sigmoid
            float inv = 1.0f / (1.0f + e);
            float p   = (x >= 0.0f) ? inv : e * inv;  // sigmoid(x), stable
            float bce = fmaxf(x, 0.0f) - x * t + __logf(1.0f + e);
            a_bce += bce;
            a_p   += p;
            a_t   += t;
            a_pt  += p * t;
            a_x   += x;
        }
    }

    // wave32 reductions via v_wmma_f32_16x16x4_f32 (EXEC all-ones here)
    float r_bce = wave_reduce_wmma(a_bce);
    float r_p   = wave_reduce_wmma(a_p);
    float r_t   = wave_reduce_wmma(a_t);
    float r_pt  = wave_reduce_wmma(a_pt);
    float r_x   = wave_reduce_wmma(a_x);

    __shared__ float sred[(THREADS / 32) * 5];
    const int lane = threadIdx.x & 31;
    const int wid  = threadIdx.x >> 5;
    if (lane == 0) {
        sred[wid * 5 + 0] = r_bce;
        sred[wid * 5 + 1] = r_p;
        sred[wid * 5 + 2] = r_t;
        sred[wid * 5 + 3] = r_pt;
        sred[wid * 5 + 4] = r_x;
    }
    __syncthreads();
    if (threadIdx.x == 0) {
        float tb = 0.0f, tp = 0.0f, tt = 0.0f, tpt = 0.0f, tx = 0.0f;
#pragma unroll
        for (int w = 0; w < THREADS / 32; ++w) {
            tb  += sred[w * 5 + 0];
            tp  += sred[w * 5 + 1];
            tt  += sred[w * 5 + 2];
            tpt += sred[w * 5 + 3];
            tx  += sred[w * 5 + 4];
        }
        atomicAdd(&ws[WS_BCE],        tb);
        atomicAdd(&ws[WS_P  + b],     tp);
        atomicAdd(&ws[WS_T  + b],     tt);
        atomicAdd(&ws[WS_PT + b],     tpt);
        atomicAdd(&ws[WS_S  + frame], tx);
    }
}

__global__ void finalize_kernel(const float* __restrict__ ws, float* __restrict__ out) {
    if (threadIdx.x == 0 && blockIdx.x == 0) {
        float bce_mean = ws[WS_BCE] * (float)(1.0 / NTOT);
        float dsum = 0.0f;
#pragma unroll
        for (int b = 0; b < BB; ++b) {
            float pt = ws[WS_PT + b];
            float p  = ws[WS_P  + b];
            float t  = ws[WS_T  + b];
            dsum += (2.0f * pt + 1e-5f) / (p + t + 1e-5f);
        }
        float dice      = 1.0f - dsum / (float)BB;
        float dice_loss = 0.5f * bce_mean + dice;
        float ssl = 0.0f;
        for (int b = 0; b < BB; ++b) {
            for (int t = 0; t < TT - 1; ++t) {
                float d = fabsf(ws[WS_S + b * TT + t + 1] - ws[WS_S + b * TT + t]);
                float r = fmaxf(d - 100.0f, 0.0f);
                ssl += r * r;
            }
        }
        out[0] = dice_loss + 0.1f * ssl;
    }
}

extern "C" void kernel_launch(void* const* d_in, const int* in_sizes, int n_in,
                              void* d_out, int out_size, void* d_ws, size_t ws_size,
                              hipStream_t stream) {
    const f32x4* in4 = (const f32x4*)d_in[0];
    const f32x4* tg4 = (const f32x4*)d_in[1];
    float* ws  = (float*)d_ws;
    float* out = (float*)d_out;

    zero_ws_kernel<<<1, THREADS, 0, stream>>>(ws);
    bce_dice_partial_kernel<<<NFRAMES * BLOCKS_PER_FRAME, THREADS, 0, stream>>>(in4, tg4, ws);
    finalize_kernel<<<1, 32, 0, stream>>>(ws, out);
}